// TransINTModel_42520176230877
// MI455X (gfx1250) — compile-verified
//
#include <hip/hip_runtime.h>

typedef __attribute__((ext_vector_type(2))) float v2f;
typedef __attribute__((ext_vector_type(8))) float v8f;

#define DIMS 256  // embedding dim (fixed by the reference problem)

// ds_swizzle group-of-32 XOR pattern: offset = xor<<10 | or<<5 | and(0x1f)
#define SWZ_XOR(x, m) \
  __int_as_float(__builtin_amdgcn_ds_swizzle(__float_as_int(x), (((m) << 10) | 0x1f)))

__device__ __forceinline__ float dot4(float4 a, float4 b) {
  return a.x * b.x + a.y * b.y + a.z * b.z + a.w * b.w;
}

// ---------------------------------------------------------------------------
// Reduce three per-lane f32 values across the wave using the matrix pipe.
// WMMA#1: A[m,:] = (p0[m], p1[m], p0[m+16], p1[m+16]); selector B puts
//   sum(p0) in column 0, sum(p1) in column 1.
// WMMA#2 accumulates on WMMA#1's D: A=(p2,0), selector column 2 -> sum(p2).
// D layout: VGPR v, lanes 0-15 = (M=v, N=lane); lanes 16-31 = (M=v+8, N=lane-16).
// Tree-summing the 8 accumulator VGPRs + one SWAPX16 swizzle leaves the full
// column sums in lanes 0/1/2; v_readlane broadcasts them as uniform SGPRs so
// the epilogue runs on the scalar float pipe.
// ---------------------------------------------------------------------------
__device__ __forceinline__ void wave_reduce3(float p0, float p1, float p2, int lane,
                                             float& r0, float& r1, float& r2) {
  v2f A1;
  A1.x = p0;
  A1.y = p1;
  v2f B1;
  B1.x = (lane == 0 || lane == 16) ? 1.0f : 0.0f;  // B rows K=0,K=2 -> col 0
  B1.y = (lane == 1 || lane == 17) ? 1.0f : 0.0f;  // B rows K=1,K=3 -> col 1
  v8f Cz = {};
  v8f D1 = __builtin_amdgcn_wmma_f32_16x16x4_f32(false, A1, false, B1, (short)0, Cz,
                                                 false, false);
  v2f A2;
  A2.x = p2;
  A2.y = 0.0f;
  v2f B2;
  B2.x = (lane == 2 || lane == 18) ? 1.0f : 0.0f;  // -> col 2
  B2.y = 0.0f;
  v8f D2 = __builtin_amdgcn_wmma_f32_16x16x4_f32(false, A2, false, B2, (short)0, D1,
                                                 false, false);
  float t = ((D2[0] + D2[1]) + (D2[2] + D2[3])) + ((D2[4] + D2[5]) + (D2[6] + D2[7]));
  t += SWZ_XOR(t, 16);  // fold M=8..15 halves (lanes 16-31) onto lanes 0-15
  r0 = __int_as_float(__builtin_amdgcn_readlane(__float_as_int(t), 0));
  r1 = __int_as_float(__builtin_amdgcn_readlane(__float_as_int(t), 1));
  r2 = __int_as_float(__builtin_amdgcn_readlane(__float_as_int(t), 2));
}

// ---------------------------------------------------------------------------
// Kernel 1: per-relation 2x2 Gram matrix + (pseudo)inverse.
// One wave32 per relation. Writes rws[r*8] = {a, b, d, m00, m01, m11, 0, 0}
// ---------------------------------------------------------------------------
__global__ void __launch_bounds__(256)
transint_relprep(const float* __restrict__ bases, float* __restrict__ rws, int R) {
  int wave = __builtin_amdgcn_readfirstlane(
      (int)((blockIdx.x * blockDim.x + threadIdx.x) >> 5));
  int lane = (int)(threadIdx.x & 31);
  if (wave >= R) return;
  const float* base0 = bases + (size_t)wave * 2 * DIMS + lane * 8;
  const float* base1 = base0 + DIMS;
  float4 x0 = ((const float4*)base0)[0];
  float4 x1 = ((const float4*)base0)[1];
  float4 y0 = ((const float4*)base1)[0];
  float4 y1 = ((const float4*)base1)[1];
  float pa = dot4(x0, x0) + dot4(x1, x1);
  float pb = dot4(x0, y0) + dot4(x1, y1);
  float pd = dot4(y0, y0) + dot4(y1, y1);

  float aa, bb, dd;
  wave_reduce3(pa, pb, pd, lane, aa, bb, dd);

  if (lane == 0) {
    float det = aa * dd - bb * bb;
    float m00, m01, m11;
    if (det != 0.0f) {
      float inv = 1.0f / det;
      m00 = dd * inv;
      m01 = -bb * inv;
      m11 = aa * inv;
    } else {  // reference fallback: inv = I / a
      m00 = 1.0f / aa;
      m01 = 0.0f;
      m11 = 1.0f / aa;
    }
    float* o = rws + (size_t)wave * 8;
    o[0] = aa; o[1] = bb; o[2] = dd;
    o[3] = m00; o[4] = m01; o[5] = m11;
    o[6] = 0.0f; o[7] = 0.0f;
  }
}

// ---------------------------------------------------------------------------
// Kernel 2: one wave32 per score element (waves [0,B) = pos, [B,2B) = neg).
// Lane l owns floats [8l, 8l+8) of each 256-float row: two b128 loads per row,
// fully coalesced; ent_emb (102 MB) stays resident in the 192 MB L2 so the
// random 4-row gathers are L2 hits. Uniform indices go through readfirstlane
// -> SMEM scalar loads. score = s - 2 u.(Mu) + (Mu)^T ATA (Mu), which equals
// ||proj||^2 for both the det!=0 and det==0 branches of the reference.
// ---------------------------------------------------------------------------
__global__ void __launch_bounds__(256)
transint_score(const int* __restrict__ pos_h, const int* __restrict__ pos_t,
               const int* __restrict__ pos_r, const int* __restrict__ neg_h,
               const int* __restrict__ neg_t, const float* __restrict__ ent_emb,
               const float* __restrict__ heads, const float* __restrict__ bases,
               const int* __restrict__ rel2head, const float* __restrict__ rel2mult,
               const float* __restrict__ rws, float* __restrict__ out, int Bn) {
  int wave = __builtin_amdgcn_readfirstlane(
      (int)((blockIdx.x * blockDim.x + threadIdx.x) >> 5));
  int lane = (int)(threadIdx.x & 31);
  if (wave >= 2 * Bn) return;  // wave-uniform; active waves keep EXEC all-ones

  bool is_pos = wave < Bn;
  int b = is_pos ? wave : wave - Bn;
  int h = (is_pos ? pos_h : neg_h)[b];  // uniform -> scalar loads
  int t = (is_pos ? pos_t : neg_t)[b];
  int r = pos_r[b];  // reference uses pos_r for BOTH pos and neg (neg_r unused)
  int k = rel2head[r];
  float mult = rel2mult[r];

  int off = lane * 8;
  const float4* eh = (const float4*)(ent_emb + (size_t)h * DIMS + off);
  const float4* et = (const float4*)(ent_emb + (size_t)t * DIMS + off);
  const float4* hd = (const float4*)(heads + (size_t)k * DIMS + off);
  const float4* a0 = (const float4*)(bases + (size_t)r * 2 * DIMS + off);
  const float4* a1 = (const float4*)(bases + (size_t)r * 2 * DIMS + DIMS + off);

  float4 eh0 = eh[0], eh1 = eh[1];
  float4 et0 = et[0], et1 = et[1];
  float4 hd0 = hd[0], hd1 = hd[1];
  float4 A00 = a0[0], A01 = a0[1];
  float4 A10 = a1[0], A11 = a1[1];

  float4 s0, s1;  // sub = e_h + mult*head - e_t
  s0.x = eh0.x + mult * hd0.x - et0.x;
  s0.y = eh0.y + mult * hd0.y - et0.y;
  s0.z = eh0.z + mult * hd0.z - et0.z;
  s0.w = eh0.w + mult * hd0.w - et0.w;
  s1.x = eh1.x + mult * hd1.x - et1.x;
  s1.y = eh1.y + mult * hd1.y - et1.y;
  s1.z = eh1.z + mult * hd1.z - et1.z;
  s1.w = eh1.w + mult * hd1.w - et1.w;

  float p0 = dot4(s0, A00) + dot4(s1, A01);  // sub . A0  (partial)
  float p1 = dot4(s0, A10) + dot4(s1, A11);  // sub . A1  (partial)
  float p2 = dot4(s0, s0) + dot4(s1, s1);    // sub . sub (partial)

  float u0, u1, ss;
  wave_reduce3(p0, p1, p2, lane, u0, u1, ss);

  // --- epilogue: score = s - 2 u.(Mu) + (Mu)^T ATA (Mu) -------------------
  const float* rw = rws + (size_t)r * 8;  // uniform -> scalar loads
  float ga = rw[0], gb = rw[1], gd = rw[2];
  float m00 = rw[3], m01 = rw[4], m11 = rw[5];
  float w0 = m00 * u0 + m01 * u1;
  float w1 = m01 * u0 + m11 * u1;
  float score = ss - 2.0f * (u0 * w0 + u1 * w1) +
                (ga * w0 * w0 + 2.0f * gb * w0 * w1 + gd * w1 * w1);

  if (lane == 0) out[wave] = score;
}

// ---------------------------------------------------------------------------
extern "C" void kernel_launch(void* const* d_in, const int* in_sizes, int n_in,
                              void* d_out, int out_size, void* d_ws, size_t ws_size,
                              hipStream_t stream) {
  const int*   pos_h    = (const int*)d_in[0];
  const int*   pos_t    = (const int*)d_in[1];
  const int*   pos_r    = (const int*)d_in[2];
  const int*   neg_h    = (const int*)d_in[3];
  const int*   neg_t    = (const int*)d_in[4];
  // d_in[5] = neg_r : unused by the reference math
  const float* ent_emb  = (const float*)d_in[6];
  const float* heads    = (const float*)d_in[7];
  const float* bases    = (const float*)d_in[8];
  const int*   rel2head = (const int*)d_in[9];
  const float* rel2mult = (const float*)d_in[10];
  float* out = (float*)d_out;
  float* rws = (float*)d_ws;  // R * 8 floats (32 KB for R=1000)

  int Bn = in_sizes[0];
  int R  = in_sizes[9];

  int relBlocks = (R * 32 + 255) / 256;
  transint_relprep<<<relBlocks, 256, 0, stream>>>(bases, rws, R);

  long nwaves = 2L * (long)Bn;
  long blocks = (nwaves * 32 + 255) / 256;
  transint_score<<<(int)blocks, 256, 0, stream>>>(
      pos_h, pos_t, pos_r, neg_h, neg_t, ent_emb, heads, bases,
      rel2head, rel2mult, rws, out, Bn);
}